// Encoder_48524540511031
// MI455X (gfx1250) — compile-verified
//
#include <hip/hip_runtime.h>
#include <hip/hip_bf16.h>

typedef __attribute__((ext_vector_type(2))) float v2f;
typedef __attribute__((ext_vector_type(8))) float v8f;

#define HDIM 128
#define GNUM 64
#define LDS_STRIDE 132   // 128 + 4 pad: lanes hit banks 4 apart -> conflict free
#define BN_EPS 1e-5f

// ---------------------------------------------------------------- utilities
__global__ void fill_f32(float* __restrict__ p, float v, int n) {
    int i = blockIdx.x * blockDim.x + threadIdx.x;
    if (i < n) p[i] = v;
}

// ---------------------------------------------------------------- degree
__global__ void count_deg(const int* __restrict__ edst, float* __restrict__ deg, int E) {
    int e = blockIdx.x * blockDim.x + threadIdx.x;
    if (e < E) atomicAdd(&deg[edst[e]], 1.0f);
}

__global__ void make_invdeg(float* __restrict__ deg, int n) {
    int i = blockIdx.x * blockDim.x + threadIdx.x;
    if (i < n) deg[i] = 1.0f / fmaxf(deg[i], 1.0f);
}

// ------------------------------------------------- edge scatter (segment sum)
// one wave (32 lanes) per edge; each lane moves 4 consecutive floats
__global__ __launch_bounds__(256)
void edge_scatter(const float* __restrict__ h, const int* __restrict__ esrc,
                  const int* __restrict__ edst, float* __restrict__ pooled, int E) {
    int e = blockIdx.x * (blockDim.x >> 5) + (threadIdx.x >> 5);
    if (e >= E) return;
    int lane = threadIdx.x & 31;
    int s = esrc[e];
    int d = edst[e];
    const float4 v = *(const float4*)(h + (size_t)s * HDIM + lane * 4);
    float* p = pooled + (size_t)d * HDIM + lane * 4;
    atomicAdd(p + 0, v.x);
    atomicAdd(p + 1, v.y);
    atomicAdd(p + 2, v.z);
    atomicAdd(p + 3, v.w);
}

// ------------------------------------------------------------- WMMA GEMM 128
// C[rows x 128] = preop(A[rows x 128]) @ W[128 x 128] + bias
// MODE 0: preop = row scale (p0 = invdeg per row)   [pooled / deg]
// MODE 1: preop = relu(p0[col]*x + p1[col])          [fused BN+ReLU input]
// In-place (Cout == Ain) is safe: A strip fully staged to LDS before stores.
template <int MODE>
__global__ __launch_bounds__(256)
void gemm128_wmma(const float* __restrict__ Ain, float* __restrict__ Cout,
                  const float* __restrict__ W, const float* __restrict__ bias,
                  const float* __restrict__ p0, const float* __restrict__ p1,
                  int nrows) {
    __shared__ float lds[16 * LDS_STRIDE];
    const int tid  = threadIdx.x;
    const int row0 = blockIdx.x * 16;
    const bool full_tile = (row0 + 16 <= nrows);

    // ---- stage 16x128 A strip with fused pre-op (2048 elems / 256 thr = 8 ea)
    {
        int e0 = tid * 8;
        int r  = e0 >> 7;
        int c0 = e0 & 127;
        int rg = row0 + r;
        if (full_tile) {
            const float* src = Ain + (size_t)rg * HDIM + c0;
            float rs = (MODE == 0) ? p0[rg] : 1.0f;
#pragma unroll
            for (int j = 0; j < 8; ++j) {
                float v = src[j];
                if (MODE == 0) v *= rs;
                else           v = fmaxf(p0[c0 + j] * v + p1[c0 + j], 0.0f);
                lds[r * LDS_STRIDE + c0 + j] = v;
            }
        } else {
            bool ok = (rg < nrows);
            const float* src = Ain + (size_t)(ok ? rg : 0) * HDIM + c0;
            float rs = (MODE == 0) ? p0[ok ? rg : 0] : 1.0f;
#pragma unroll
            for (int j = 0; j < 8; ++j) {
                float v = ok ? src[j] : 0.0f;
                if (MODE == 0) v *= rs;
                else           v = fmaxf(p0[c0 + j] * v + p1[c0 + j], 0.0f);
                lds[r * LDS_STRIDE + c0 + j] = v;
            }
        }
    }
    __syncthreads();

    const int wave = tid >> 5;       // 8 waves -> 8 column tiles
    const int lane = tid & 31;
    const int half = lane >> 4;      // K sub-pair select (ISA 16x4 f32 layout)
    const int mlo  = lane & 15;
    const int n0   = wave * 16;
    const int ncol = n0 + mlo;

    v8f acc = {};
#pragma unroll 8
    for (int k = 0; k < HDIM; k += 4) {
        int ka = k + 2 * half;
        v2f a, b;
        a.x = lds[mlo * LDS_STRIDE + ka];
        a.y = lds[mlo * LDS_STRIDE + ka + 1];
        b.x = W[ka * HDIM + ncol];
        b.y = W[(ka + 1) * HDIM + ncol];
        acc = __builtin_amdgcn_wmma_f32_16x16x4_f32(false, a, false, b,
                                                    (short)0, acc, false, false);
    }

    const float bn = bias[ncol];
    if (full_tile) {
        // branch-free epilogue: no per-element exec masking
        float* cb = Cout + (size_t)(row0 + 8 * half) * HDIM + ncol;
#pragma unroll
        for (int j = 0; j < 8; ++j)
            cb[(size_t)j * HDIM] = acc[j] + bn;
    } else {
#pragma unroll
        for (int j = 0; j < 8; ++j) {
            int m = row0 + j + 8 * half;   // C/D layout: VGPR j -> M = j + 8*half
            if (m < nrows) Cout[(size_t)m * HDIM + ncol] = acc[j] + bn;
        }
    }
}

// ------------------------------------------------------------ BN column stats
__global__ void col_stats(const float* __restrict__ T, float* __restrict__ sums,
                          float* __restrict__ sumsq, int nrows) {
    int c = threadIdx.x;              // 128 threads = 128 columns
    float s = 0.0f, sq = 0.0f;
    for (int r = blockIdx.x; r < nrows; r += gridDim.x) {
        float v = T[(size_t)r * HDIM + c];
        s += v;
        sq += v * v;
    }
    atomicAdd(&sums[c], s);
    atomicAdd(&sumsq[c], sq);
}

__global__ void bn_finalize(const float* __restrict__ sums, const float* __restrict__ sumsq,
                            const float* __restrict__ gamma, const float* __restrict__ beta,
                            float* __restrict__ scaleA, float* __restrict__ shiftB,
                            float inv_n) {
    int c = threadIdx.x;
    float mu  = sums[c] * inv_n;
    float var = fmaxf(sumsq[c] * inv_n - mu * mu, 0.0f);
    float a   = gamma[c] * rsqrtf(var + BN_EPS);
    scaleA[c] = a;
    shiftB[c] = beta[c] - a * mu;
}

__global__ void apply_affine_relu(const float* __restrict__ in, float* __restrict__ out,
                                  const float* __restrict__ a, const float* __restrict__ b,
                                  int n) {
    int i = blockIdx.x * blockDim.x + threadIdx.x;
    if (i < n) {
        int c = i & (HDIM - 1);
        out[i] = fmaxf(a[c] * in[i] + b[c], 0.0f);
    }
}

// ------------------------------------------------------------- graph pooling
__global__ void count_graphs(const int* __restrict__ gid, float* __restrict__ cnt, int n) {
    int i = blockIdx.x * blockDim.x + threadIdx.x;
    if (i < n) atomicAdd(&cnt[gid[i]], 1.0f);
}

__global__ __launch_bounds__(256)
void node_pool_scatter(const float* __restrict__ h, const int* __restrict__ gid,
                       float* __restrict__ gpool, int n) {
    int node = blockIdx.x * (blockDim.x >> 5) + (threadIdx.x >> 5);
    if (node >= n) return;
    int lane = threadIdx.x & 31;
    int g = gid[node];
    const float4 v = *(const float4*)(h + (size_t)node * HDIM + lane * 4);
    float* p = gpool + (size_t)g * HDIM + lane * 4;
    atomicAdd(p + 0, v.x);
    atomicAdd(p + 1, v.y);
    atomicAdd(p + 2, v.z);
    atomicAdd(p + 3, v.w);
}

__global__ void pool_divide(float* __restrict__ gpool, const float* __restrict__ cnt, int n) {
    int i = blockIdx.x * blockDim.x + threadIdx.x;
    if (i < n) gpool[i] /= fmaxf(cnt[i >> 7], 1.0f);
}

// ------------------------------------------------------------------- launcher
extern "C" void kernel_launch(void* const* d_in, const int* in_sizes, int n_in,
                              void* d_out, int out_size, void* d_ws, size_t ws_size,
                              hipStream_t stream) {
    const float* x    = (const float*)d_in[0];
    const int*   esrc = (const int*)d_in[1];
    const int*   edst = (const int*)d_in[2];
    const int*   gid  = (const int*)d_in[3];
    const float* W1   = (const float*)d_in[4];
    const float* b1   = (const float*)d_in[5];
    const float* g1   = (const float*)d_in[6];
    const float* be1  = (const float*)d_in[7];
    const float* W2   = (const float*)d_in[8];
    const float* b2   = (const float*)d_in[9];
    const float* gbn  = (const float*)d_in[10];
    const float* bbn  = (const float*)d_in[11];

    const int N = in_sizes[3];        // graph_id length
    const int E = in_sizes[1];
    const int L = 2;
    const int NH = N * HDIM;

    float* out      = (float*)d_out;
    float* out_pool = out;                 // [G, H]
    float* out_h    = out + GNUM * HDIM;   // [N, H]

    // ---- workspace carving (floats)
    float* ws     = (float*)d_ws;
    float* hbuf   = ws;              ws += NH;   // intermediate h
    float* pooled = ws;              ws += NH;   // pooled / t1 / t3 (in-place GEMMs)
    float* invdeg = ws;              ws += N;
    float* sums   = ws;              ws += HDIM;
    float* sumsq  = ws;              ws += HDIM;
    float* scaleA = ws;              ws += HDIM;
    float* shiftB = ws;              ws += HDIM;
    float* cnt    = ws;              ws += GNUM;

    const int TB = 256;
    const int rowTiles = (N + 15) / 16;
    const int edgeWaveBlocks = (E + 7) / 8;
    const int nodeWaveBlocks = (N + 7) / 8;
    const float inv_n = 1.0f / (float)N;

    // ---- degree -> 1/max(deg,1)
    fill_f32<<<(N + TB - 1) / TB, TB, 0, stream>>>(invdeg, 0.0f, N);
    count_deg<<<(E + TB - 1) / TB, TB, 0, stream>>>(edst, invdeg, E);
    make_invdeg<<<(N + TB - 1) / TB, TB, 0, stream>>>(invdeg, N);

    const float* h_src = x;
    for (int l = 0; l < L; ++l) {
        float* h_dst = (l == L - 1) ? out_h : hbuf;

        // segment_sum(h[edge_src], edge_dst)
        fill_f32<<<(NH + TB - 1) / TB, TB, 0, stream>>>(pooled, 0.0f, NH);
        edge_scatter<<<edgeWaveBlocks, TB, 0, stream>>>(h_src, esrc, edst, pooled, E);

        // t1 = (pooled/deg) @ W1[l] + b1[l]   (in-place, row-scale fused)
        gemm128_wmma<0><<<rowTiles, TB, 0, stream>>>(
            pooled, pooled, W1 + (size_t)l * HDIM * HDIM, b1 + l * HDIM,
            invdeg, nullptr, N);

        // BN1 stats + affine
        fill_f32<<<1, 2 * HDIM, 0, stream>>>(sums, 0.0f, 2 * HDIM);
        col_stats<<<512, HDIM, 0, stream>>>(pooled, sums, sumsq, N);
        bn_finalize<<<1, HDIM, 0, stream>>>(sums, sumsq, g1 + l * HDIM, be1 + l * HDIM,
                                            scaleA, shiftB, inv_n);

        // t3 = relu(BN1(t1)) @ W2[l] + b2[l]  (in-place, BN+ReLU fused into staging)
        gemm128_wmma<1><<<rowTiles, TB, 0, stream>>>(
            pooled, pooled, W2 + (size_t)l * HDIM * HDIM, b2 + l * HDIM,
            scaleA, shiftB, N);

        // outer BN stats + affine, then h = relu(BN2(t3))
        fill_f32<<<1, 2 * HDIM, 0, stream>>>(sums, 0.0f, 2 * HDIM);
        col_stats<<<512, HDIM, 0, stream>>>(pooled, sums, sumsq, N);
        bn_finalize<<<1, HDIM, 0, stream>>>(sums, sumsq, gbn + l * HDIM, bbn + l * HDIM,
                                            scaleA, shiftB, inv_n);
        apply_affine_relu<<<(NH + TB - 1) / TB, TB, 0, stream>>>(pooled, h_dst,
                                                                 scaleA, shiftB, NH);
        h_src = h_dst;
    }

    // ---- per-graph mean pooling into d_out[0 : G*H]
    fill_f32<<<1, GNUM, 0, stream>>>(cnt, 0.0f, GNUM);
    count_graphs<<<(N + TB - 1) / TB, TB, 0, stream>>>(gid, cnt, N);
    fill_f32<<<(GNUM * HDIM + TB - 1) / TB, TB, 0, stream>>>(out_pool, 0.0f, GNUM * HDIM);
    node_pool_scatter<<<nodeWaveBlocks, TB, 0, stream>>>(out_h, gid, out_pool, N);
    pool_divide<<<(GNUM * HDIM + TB - 1) / TB, TB, 0, stream>>>(out_pool, cnt, GNUM * HDIM);
}